// FrequencyEnergyIndicator_89816356094317
// MI455X (gfx1250) — compile-verified
//
#include <hip/hip_runtime.h>

typedef float v2f __attribute__((ext_vector_type(2)));
typedef float v8f __attribute__((ext_vector_type(8)));

#define IMG_HW  128
#define IMG_PIX (128 * 128)
#define XW      33          // padded LDS row stride for 32-wide halo tile
#define HWS     18          // padded LDS row stride for 16-wide H buffer
#define NWAVE   8
#define WPAD    20          // zero-pad offset in the banded-weight tables

static __device__ __forceinline__ v8f wmma4(v2f a, v2f b, v8f c) {
  // D(16x16,f32) = A(16x4,f32) * B(4x16,f32) + C
  return __builtin_amdgcn_wmma_f32_16x16x4_f32(
      /*neg_a=*/false, a, /*neg_b=*/false, b,
      /*c_mod=*/(short)0, c, /*reuse_a=*/false, /*reuse_b=*/false);
}

// One wave (wave32) computes one 16x16 output tile of one (batch,channel) image.
// Separable Gaussian as banded matmuls on the WMMA pipe:
//   H(32x16) = X(32x32) * Wh(32x16)   (horizontal, incl. vertical halo rows)
//   V(16x16) = Wv(16x32) * H(32x16)   (vertical)
// Wh[k][j] = Wv[j][k] = w[k - j - (8-pad)]  -> one fragment set serves both.
__global__ __launch_bounds__(256) void bands_kernel(const float* __restrict__ z,
                                                    float* __restrict__ out) {
  __shared__ float sWp[2][64];             // zero-padded banded-weight tables
  __shared__ float sX[NWAVE][32 * XW];     // 32x32 halo tiles
  __shared__ float sH[NWAVE][32 * HWS];    // 32x16 horizontal-pass results

  const int tid = threadIdx.x;

  // Zero the padded tables, then write normalized 1-D kernels (2-D Gaussian
  // factors exactly into normalized 1-D outer product).
  if (tid < 128) ((float*)sWp)[tid] = 0.f;
  __syncthreads();
  if (tid == 0) {          // blur2: sigma=2, 17 taps, boff = 8-pad = 0
    float e[17]; float s = 0.f;
#pragma unroll
    for (int t = 0; t < 17; ++t) { float d = (float)t - 8.f; e[t] = __expf(-0.125f * d * d); s += e[t]; }
#pragma unroll
    for (int t = 0; t < 17; ++t) sWp[0][WPAD + t] = e[t] / s;
  } else if (tid == 1) {   // blur1: sigma=1, 9 taps, boff = 8-pad = 4
    float e[9]; float s = 0.f;
#pragma unroll
    for (int t = 0; t < 9; ++t) { float d = (float)t - 4.f; e[t] = __expf(-0.5f * d * d); s += e[t]; }
#pragma unroll
    for (int t = 0; t < 9; ++t) sWp[1][WPAD + 4 + t] = e[t] / s;
  }
  __syncthreads();   // last block-wide barrier: waves are independent below

  const int wave = tid >> 5;
  const int lane = tid & 31;
  const int colN = lane & 15;   // N column (B/C frag) == M row (A frag)
  const int hf   = lane >> 4;   // half-wave select (K sub-pair / M+8 rows)

  const long tileId = (long)blockIdx.x * NWAVE + wave;   // 131072 tiles total
  const int  tIn = (int)(tileId & 63);
  const long img = tileId >> 6;                          // batch*256 + channel
  const int  r0 = (tIn >> 3) * 16;
  const int  c0 = (tIn & 7) * 16;

  const float* __restrict__ src = z + img * (long)IMG_PIX;
  float* __restrict__ X = &sX[wave][0];
  float* __restrict__ H = &sH[wave][0];

  // Stage 32x32 zero-padded halo tile (coalesced: lanes span columns).
  {
    const int  gc  = c0 - 8 + lane;
    const bool cok = (unsigned)gc < (unsigned)IMG_HW;
#pragma unroll 4
    for (int r = 0; r < 32; ++r) {
      const int gr = r0 - 8 + r;
      float v = 0.f;
      if (cok && (unsigned)gr < (unsigned)IMG_HW) v = src[gr * IMG_HW + gc];
      X[r * XW + lane] = v;
    }
  }
  __builtin_amdgcn_wave_barrier();   // LDS is in-order within a wave

  v8f V[2];   // V[0] = blur(sigma=2), V[1] = blur(sigma=1)

#pragma unroll
  for (int p = 0; p < 2; ++p) {
    const float* __restrict__ Wp = &sWp[p][0];

    // Banded-weight fragments: shared by horizontal-B and vertical-A roles.
    v2f wf[8];
#pragma unroll
    for (int kc = 0; kc < 8; ++kc) {
      const int d = kc * 4 + hf * 2 - colN + WPAD;   // in [5, 51] -> no predication
      wf[kc].x = Wp[d];
      wf[kc].y = Wp[d + 1];
    }

    // ---- horizontal pass: two 16-row blocks -> H[32][16] in LDS ----
#pragma unroll
    for (int m = 0; m < 2; ++m) {
      v8f acc = {};
#pragma unroll
      for (int kc = 0; kc < 8; ++kc) {
        const int kk = kc * 4 + hf * 2;
        v2f a;
        a.x = X[(m * 16 + colN) * XW + kk];        // A: X rows, K cols
        a.y = X[(m * 16 + colN) * XW + kk + 1];
        acc = wmma4(a, wf[kc], acc);
      }
#pragma unroll
      for (int v = 0; v < 8; ++v)
        H[(m * 16 + v + 8 * hf) * HWS + colN] = acc[v];   // C-layout -> LDS
    }
    __builtin_amdgcn_wave_barrier();

    // ---- vertical pass: V = Wv(16x32) * H(32x16) ----
    {
      v8f acc = {};
#pragma unroll
      for (int kc = 0; kc < 8; ++kc) {
        const int kk = kc * 4 + hf * 2;
        v2f b;
        b.x = H[kk * HWS + colN];                  // B: H rows striped over lanes
        b.y = H[(kk + 1) * HWS + colN];
        acc = wmma4(wf[kc], b, acc);
      }
      V[p] = acc;
    }
    __builtin_amdgcn_wave_barrier();   // H reused by next blur (same wave)
  }

  // ---- bands: [blur2, blur1-blur2, z-blur1], layout (3,8,256,128,128) ----
  const long BSTRIDE = 33554432L;  // 8*256*128*128
  float* __restrict__ ob = out + 24;
  const long ibase = img * (long)IMG_PIX;
#pragma unroll
  for (int v = 0; v < 8; ++v) {
    const int  grow = r0 + v + 8 * hf;
    const long pidx = ibase + (long)grow * IMG_HW + (c0 + colN);
    const float xc  = X[(v + 8 * hf + 8) * XW + (colN + 8)];  // center sample
    const float b2v = V[0][v];
    const float b1v = V[1][v];
    ob[pidx]               = b2v;
    ob[BSTRIDE + pidx]     = b1v - b2v;
    ob[2 * BSTRIDE + pidx] = xc - b1v;
  }
}

// Deterministic energy reduction, stage 1: 32768-element chunks -> partial sums.
// Chunks never straddle a (band,batch) slab (4194304 % 32768 == 0).
__global__ __launch_bounds__(256) void energy_partial(const float* __restrict__ bands,
                                                      float* __restrict__ partial) {
  const long base = (long)blockIdx.x * 32768L;
  float s = 0.f;
  for (int i = threadIdx.x; i < 32768; i += 256) {
    const float x = bands[base + i];
    s += x * x;
  }
  __shared__ float red[256];
  red[threadIdx.x] = s;
  __syncthreads();
#pragma unroll
  for (int off = 128; off > 0; off >>= 1) {
    if (threadIdx.x < off) red[threadIdx.x] += red[threadIdx.x + off];
    __syncthreads();
  }
  if (threadIdx.x == 0) partial[blockIdx.x] = red[0];
}

// Stage 2: fold 128 partials per (band,batch), normalize per batch -> e_t (8x3).
__global__ void finalize_energy(const float* __restrict__ partial,
                                float* __restrict__ out) {
  __shared__ float E[24];          // index = band*8 + batch
  const int t = threadIdx.x;
  if (t < 24) {
    float s = 0.f;
    for (int i = 0; i < 128; ++i) s += partial[t * 128 + i];
    E[t] = s;
  }
  __syncthreads();
  if (t < 24) {
    const int batch = t / 3, band = t % 3;
    const float denom = E[batch] + E[8 + batch] + E[16 + batch] + 1e-8f;
    out[batch * 3 + band] = E[band * 8 + batch] / denom;
  }
}

extern "C" void kernel_launch(void* const* d_in, const int* in_sizes, int n_in,
                              void* d_out, int out_size, void* d_ws, size_t ws_size,
                              hipStream_t stream) {
  (void)in_sizes; (void)n_in; (void)out_size; (void)ws_size;
  const float* z   = (const float*)d_in[0];
  float*       out = (float*)d_out;          // [0..23]=e_t, [24..)=bands_stacked
  float*       ws  = (float*)d_ws;           // 3072 partial sums (12 KB)

  // 2048 images * 64 tiles = 131072 tiles; 8 waves (tiles) per 256-thread block.
  bands_kernel<<<16384, 256, 0, stream>>>(z, out);
  // 3 bands * 8 batches * 128 chunks = 3072 blocks.
  energy_partial<<<3072, 256, 0, stream>>>(out + 24, ws);
  finalize_energy<<<1, 32, 0, stream>>>(ws, out);
}